// LSTMModel_55113020342722
// MI455X (gfx1250) — compile-verified
//
#include <hip/hip_runtime.h>
#include <math.h>

// ---------------------------------------------------------------------------
// Problem constants (from the reference): H = 4096, IN = 4, gates = 4*H rows.
// ---------------------------------------------------------------------------
#define HDIM   4096
#define ROWS   16384           // 4*H rows of W_hh / W_ih
#define KSPLIT 4               // K-dimension split per 16-row tile
#define KCHUNK (HDIM / KSPLIT) // 1024 columns per wave

typedef __attribute__((ext_vector_type(2))) float v2f;
typedef __attribute__((ext_vector_type(8))) float v8f;

// ---------------------------------------------------------------------------
// Kernel A: gates_part[slice][row] = (W_hh[row, kbase:kbase+KCHUNK] . h[...])
// One wave32 per (16-row tile, K-slice). Uses V_WMMA_F32_16X16X4_F32.
//
// A layout (ISA 7.12.2, 32-bit A 16x4): lane m<16 holds row m, VGPR0=K0,
// VGPR1=K1; lanes 16-31 hold row m-16, VGPR0=K2, VGPR1=K3.  -> one b64 load.
// B (4x16) is h[k..k+3] replicated across all 16 N columns, so every lane of
// D VGPR j holds the finished dot product of row (j + 8*laneHalf).
// ---------------------------------------------------------------------------
__global__ __launch_bounds__(256) void lstm_gates_wmma(
    const float* __restrict__ Whh,   // [ROWS, HDIM] row-major
    const float* __restrict__ h,     // [HDIM]
    float* __restrict__ gates_part)  // [KSPLIT, ROWS]
{
    const int lane  = threadIdx.x & 31;
    const int wave  = threadIdx.x >> 5;
    const int wid   = blockIdx.x * (blockDim.x >> 5) + wave; // 0..4095
    const int slice = wid & (KSPLIT - 1);
    const int tile  = wid >> 2;            // 0..1023 (16-row tile index)
    const int r0    = tile * 16;
    const int half  = lane >> 4;           // 0 => lanes 0-15, 1 => lanes 16-31
    const int m     = lane & 15;           // row within the tile
    const int kbase = slice * KCHUNK;

    // Per-lane streaming pointers (8-byte aligned: kbase%4==0, +2*half)
    const float* __restrict__ aptr =
        Whh + (size_t)(r0 + m) * HDIM + kbase + (half << 1);
    const float* __restrict__ bptr = h + kbase + (half << 1);

    v8f acc = {}; // 16x16 f32 accumulator, columns all identical

#pragma unroll 4
    for (int k = 0; k < KCHUNK; k += 4) {
        v2f a = *(const v2f*)(aptr + k);   // W rows, ISA A-layout
        v2f b = *(const v2f*)(bptr + k);   // h broadcast across N
        // (neg_a, A, neg_b, B, c_mod, C, reuse_a, reuse_b)
        acc = __builtin_amdgcn_wmma_f32_16x16x4_f32(
            false, a, false, b, (short)0, acc, false, false);
    }

    // D layout: VGPR j -> row r0+j (lanes 0-15) / row r0+8+j (lanes 16-31),
    // replicated over the 16 N-lanes.  Lanes 0 and 16 each emit 8 rows.
    if (m == 0) {
        const int rbase = r0 + (half << 3);
        float* __restrict__ dst = gates_part + (size_t)slice * ROWS + rbase;
#pragma unroll
        for (int j = 0; j < 8; ++j) dst[j] = acc[j];
    }
}

// ---------------------------------------------------------------------------
// Kernel B: combine K-slices + biases + W_ih@x, apply LSTM cell update.
// Writes h_new -> out[4 .. 4+H), c_new -> out[4+H .. 4+2H).
// ---------------------------------------------------------------------------
__device__ __forceinline__ float sigmoidf_(float v) {
    return 1.0f / (1.0f + expf(-v));
}

__global__ __launch_bounds__(256) void lstm_cell_update(
    const float* __restrict__ gates_part, // [KSPLIT, ROWS]
    const float* __restrict__ Wih,        // [ROWS, 4]
    const float* __restrict__ x,          // [4]
    const float* __restrict__ bih,        // [ROWS]
    const float* __restrict__ bhh,        // [ROWS]
    const float* __restrict__ c_in,       // [HDIM]
    float* __restrict__ out)              // [4 + 2H]
{
    const int i = blockIdx.x * blockDim.x + threadIdx.x;
    if (i >= HDIM) return;

    const float x0 = x[0], x1 = x[1], x2 = x[2], x3 = x[3];

    float g[4];
#pragma unroll
    for (int gate = 0; gate < 4; ++gate) {
        const int row = gate * HDIM + i;
        float v = bih[row] + bhh[row];
#pragma unroll
        for (int s = 0; s < KSPLIT; ++s) v += gates_part[(size_t)s * ROWS + row];
        const float* wr = Wih + (size_t)row * 4;
        v += wr[0] * x0 + wr[1] * x1 + wr[2] * x2 + wr[3] * x3;
        g[gate] = v;
    }

    const float ig = sigmoidf_(g[0]);
    const float fg = sigmoidf_(g[1]);
    const float gg = tanhf(g[2]);
    const float og = sigmoidf_(g[3]);

    const float cn = fg * c_in[i] + ig * gg;
    const float hn = og * tanhf(cn);

    out[4 + i]        = hn;   // h_new
    out[4 + HDIM + i] = cn;   // c_new
}

// ---------------------------------------------------------------------------
// Kernel C: tiny MLP head.  y = w3 @ relu(w2 @ relu(w1 @ h_new + b1) + b2) + b3
// One 256-thread block; w1 GEMV via LDS tree reduction (192 KB of w1, trivial).
// ---------------------------------------------------------------------------
__global__ __launch_bounds__(256) void mlp_head(
    const float* __restrict__ out_ro,  // d_out (h_new lives at +4)
    const float* __restrict__ w1, const float* __restrict__ b1,
    const float* __restrict__ w2, const float* __restrict__ b2,
    const float* __restrict__ w3, const float* __restrict__ b3,
    float* __restrict__ out)
{
    __shared__ float red[12][256];
    const float* __restrict__ hn = out_ro + 4;
    const int tid = threadIdx.x;

    float acc[12];
#pragma unroll
    for (int r = 0; r < 12; ++r) acc[r] = 0.0f;

    for (int col = tid; col < HDIM; col += 256) {
        const float hv = hn[col];
#pragma unroll
        for (int r = 0; r < 12; ++r) acc[r] += w1[(size_t)r * HDIM + col] * hv;
    }
#pragma unroll
    for (int r = 0; r < 12; ++r) red[r][tid] = acc[r];
    __syncthreads();

    for (int s = 128; s > 0; s >>= 1) {
        if (tid < s) {
#pragma unroll
            for (int r = 0; r < 12; ++r) red[r][tid] += red[r][tid + s];
        }
        __syncthreads();
    }

    if (tid == 0) {
        float y1[12], y2[8];
#pragma unroll
        for (int r = 0; r < 12; ++r) {
            float v = red[r][0] + b1[r];
            y1[r] = v > 0.0f ? v : 0.0f;
        }
#pragma unroll
        for (int j = 0; j < 8; ++j) {
            float v = b2[j];
#pragma unroll
            for (int r = 0; r < 12; ++r) v += w2[j * 12 + r] * y1[r];
            y2[j] = v > 0.0f ? v : 0.0f;
        }
#pragma unroll
        for (int q = 0; q < 4; ++q) {
            float v = b3[q];
#pragma unroll
            for (int j = 0; j < 8; ++j) v += w3[q * 8 + j] * y2[j];
            out[q] = v;
        }
    }
}

// ---------------------------------------------------------------------------
// Launch. Input order (setup_inputs dict): input, hidden, cell, W_ih, W_hh,
// b_ih, b_hh, w1, b1, w2, b2, w3, b3.  Output: [output(4), h_new(H), c_new(H)].
// Scratch: KSPLIT * ROWS floats = 256 KB of partial gate sums.
// ---------------------------------------------------------------------------
extern "C" void kernel_launch(void* const* d_in, const int* in_sizes, int n_in,
                              void* d_out, int out_size, void* d_ws, size_t ws_size,
                              hipStream_t stream) {
    const float* x   = (const float*)d_in[0];
    const float* h   = (const float*)d_in[1];
    const float* c   = (const float*)d_in[2];
    const float* Wih = (const float*)d_in[3];
    const float* Whh = (const float*)d_in[4];
    const float* bih = (const float*)d_in[5];
    const float* bhh = (const float*)d_in[6];
    const float* w1  = (const float*)d_in[7];
    const float* b1  = (const float*)d_in[8];
    const float* w2  = (const float*)d_in[9];
    const float* b2  = (const float*)d_in[10];
    const float* w3  = (const float*)d_in[11];
    const float* b3  = (const float*)d_in[12];

    float* out   = (float*)d_out;
    float* gpart = (float*)d_ws;   // [KSPLIT * ROWS] floats

    // 1024 row-tiles * KSPLIT waves = 4096 waves; 8 waves (256 thr) per block.
    lstm_gates_wmma<<<(1024 * KSPLIT) / 8, 256, 0, stream>>>(Whh, h, gpart);
    lstm_cell_update<<<HDIM / 256, 256, 0, stream>>>(gpart, Wih, x, bih, bhh, c, out);
    mlp_head<<<1, 256, 0, stream>>>(out, w1, b1, w2, b2, w3, b3, out);
}